// ColorQuantizer_47777216200847
// MI455X (gfx1250) — compile-verified
//
#include <hip/hip_runtime.h>

// ---------------------------------------------------------------------------
// ColorQuantizer forward for MI455X (gfx1250, wave32, WMMA).
//
// Key identity: out = stop_gradient(hard - soft) + soft  ==  hard  (forward),
// and argmax(softmax(-dist/T)) == argmin(dist) == argmax(p.c - |c|^2/2).
// So we compute: relu(x@W1+b1) -> @W2+b2 -> nearest palette color -> gather.
//
// All matmuls run on V_WMMA_F32_16X16X4_F32 with pixels along N (lane%16):
//   D1(unit,px)   = W1^T x X          (2 unit-tiles x 2 pixel-tiles = 4 wmma)
//   score(col,px) = (PAL@W2^T) x relu(D1) + (PAL.b2 - |PAL|^2/2)   (8 wmma/tile)
// Layer-2 and palette cross-term are fused into one 16x32 constant matrix M3.
// Half-lane exchanges (D-layout -> B-layout) use v_permlanex16 (VALU).
// ---------------------------------------------------------------------------

typedef float v2f __attribute__((ext_vector_type(2)));
typedef float v8f __attribute__((ext_vector_type(8)));

#define HWC 262144  // 512*512 elements per channel plane
#define PG  ((128.0f/255.0f)*2.0f - 1.0f)
#define PS  ((192.0f/255.0f)*2.0f - 1.0f)

static __device__ const float PALV[16][3] = {
    {-1.f,-1.f,-1.f}, { 1.f, 1.f, 1.f}, { 1.f,-1.f,-1.f}, {-1.f, 1.f,-1.f},
    {-1.f,-1.f, 1.f}, { 1.f, 1.f,-1.f}, { 1.f,-1.f, 1.f}, {-1.f, 1.f, 1.f},
    {  PG,  PG,  PG}, {  PG,-1.f,-1.f}, {-1.f,  PG,-1.f}, {-1.f,-1.f,  PG},
    {  PG,  PG,-1.f}, {  PG,-1.f,  PG}, {-1.f,  PG,  PG}, {  PS,  PS,  PS}
};

// Exchange data between lane-halves (lane L <-> lane L^16).
static __device__ __forceinline__ int hswap_i(int v) {
#if defined(__HIP_DEVICE_COMPILE__)
#if __has_builtin(__builtin_amdgcn_permlanex16)
    return __builtin_amdgcn_permlanex16(v, v, 0x76543210, (int)0xfedcba98,
                                        false, false);
#else
    // ds_swizzle SWAPX16: xor=0x10, or=0, and=0x1f -> offset 0x401F
    return __builtin_amdgcn_ds_swizzle(v, 0x401F);
#endif
#else
    return v;   // host pass: never codegen'd
#endif
}
static __device__ __forceinline__ float hswap_f(float x) {
    return __int_as_float(hswap_i(__float_as_int(x)));
}

static __device__ __forceinline__ v8f wmma4(v2f a, v2f b, v8f c) {
#if defined(__HIP_DEVICE_COMPILE__)
    // (neg_a, A, neg_b, B, c_mod, C, reuse_a, reuse_b)
    return __builtin_amdgcn_wmma_f32_16x16x4_f32(false, a, false, b,
                                                 (short)0, c, false, false);
#else
    return c;   // host pass placeholder, never executed
#endif
}

__global__ __launch_bounds__(256)
void cq_wmma_kernel(const float* __restrict__ x,
                    const float* __restrict__ W1,   // [3,32]
                    const float* __restrict__ b1,   // [32]
                    const float* __restrict__ W2,   // [32,3]
                    const float* __restrict__ b2,   // [3]
                    float* __restrict__ out,
                    int npix) {
    const int tid  = blockIdx.x * blockDim.x + threadIdx.x;
    const int lane = threadIdx.x & 31;
    const bool islo = lane < 16;
    const int  m    = lane & 15;

    // ---- Loop-invariant A-matrices / C-biases (per-lane init) -------------
    // A1[t]: W1^T tile, M=unit(16t..16t+15)=lane%16, K=channel:
    //   lanes 0-15 hold K=0,1; lanes 16-31 hold K=2,3 (pad).
    v2f A1[2];
#pragma unroll
    for (int t = 0; t < 2; ++t) {
        A1[t].x = islo ? W1[0*32 + 16*t + m] : W1[2*32 + 16*t + m];
        A1[t].y = islo ? W1[1*32 + 16*t + m] : 0.0f;
    }
    // C for layer 1: C(m=unit,n)=b1[unit]; VGPR i: half0 -> unit 16t+i,
    // half1 -> unit 16t+i+8.
    v8f B1C[2];
#pragma unroll
    for (int t = 0; t < 2; ++t)
#pragma unroll
        for (int i = 0; i < 8; ++i)
            B1C[t][i] = b1[16*t + i + (islo ? 0 : 8)];

    // Fused layer2+palette: M3[col][u] = sum_ch PAL[col][ch]*W2[u][ch].
    const float pm0 = PALV[m][0], pm1 = PALV[m][1], pm2 = PALV[m][2];
    v2f A3[8];
#pragma unroll
    for (int c = 0; c < 8; ++c) {
        const int k0 = 4*c + (islo ? 0 : 2);
        const int k1 = k0 + 1;
        A3[c].x = pm0*W2[k0*3+0] + pm1*W2[k0*3+1] + pm2*W2[k0*3+2];
        A3[c].y = pm0*W2[k1*3+0] + pm1*W2[k1*3+1] + pm2*W2[k1*3+2];
    }
    // Score bias: C3(col) = PAL[col].b2 - |PAL[col]|^2 / 2
    const float b20 = b2[0], b21 = b2[1], b22 = b2[2];
    v8f C3;
#pragma unroll
    for (int i = 0; i < 8; ++i) {
        const int col = i + (islo ? 0 : 8);
        const float q0 = PALV[col][0], q1 = PALV[col][1], q2 = PALV[col][2];
        C3[i] = q0*b20 + q1*b21 + q2*b22 - 0.5f*(q0*q0 + q1*q1 + q2*q2);
    }

    // Palette gather table in LDS (16B-padded rows).
    __shared__ float SP[16][4];
    if (threadIdx.x < 16) {
        SP[threadIdx.x][0] = PALV[threadIdx.x][0];
        SP[threadIdx.x][1] = PALV[threadIdx.x][1];
        SP[threadIdx.x][2] = PALV[threadIdx.x][2];
        SP[threadIdx.x][3] = 0.0f;
    }
    __syncthreads();

    const int stride = gridDim.x * blockDim.x;
    const int iters  = npix / stride;   // uniform; host guarantees exact div

#pragma unroll 1
    for (int it = 0; it < iters; ++it) {
        const int p    = tid + it * stride;
        const int base = (p >> 18) * (3*HWC) + (p & (HWC - 1));

        const float r  = __builtin_nontemporal_load(x + base);
        const float g  = __builtin_nontemporal_load(x + base + HWC);
        const float bl = __builtin_nontemporal_load(x + base + 2*HWC);

        // B1: 4(K=ch) x 16(N=px).  v0: half0=K0, half1=K2; v1: half0=K1, half1=K3.
        const float rs = hswap_f(r), gs = hswap_f(g), bs = hswap_f(bl);
        v2f B1t[2];
        B1t[0].x = islo ? r  : bs;  B1t[0].y = islo ? g  : 0.0f;  // px 0-15
        B1t[1].x = islo ? rs : bl;  B1t[1].y = islo ? gs : 0.0f;  // px 16-31

        int idxP[2];
#pragma unroll
        for (int pt = 0; pt < 2; ++pt) {
            // Layer 1 + ReLU: hidden units 0-15 (h0) and 16-31 (h1).
            v8f h0 = wmma4(A1[0], B1t[pt], B1C[0]);
            v8f h1 = wmma4(A1[1], B1t[pt], B1C[1]);
#pragma unroll
            for (int i = 0; i < 8; ++i) {
                h0[i] = fmaxf(h0[i], 0.0f);
                h1[i] = fmaxf(h1[i], 0.0f);
            }
            // score(col,px) = sum_c A3[c] x B2chunk(c) + C3
            v8f d = C3;
#pragma unroll
            for (int c = 0; c < 8; ++c) {
                const v8f& H = (c < 4) ? h0 : h1;
                const int jj = (c & 3) * 4;     // local unit base 0,4,8,12
                const int q  = jj & 4;
                const float e0 = H[q], e1 = H[q+1], e2 = H[q+2], e3 = H[q+3];
                v2f Bc;
                if (jj < 8) {   // units live in half0 of D-layout
                    Bc.x = islo ? e0 : hswap_f(e2);
                    Bc.y = islo ? e1 : hswap_f(e3);
                } else {        // units live in half1 of D-layout
                    Bc.x = islo ? hswap_f(e0) : e2;
                    Bc.y = islo ? hswap_f(e1) : e3;
                }
                d = wmma4(A3[c], Bc, d);
            }
            // Argmax over 16 colors (ties -> lowest index, as jnp.argmax).
            float best = d[0];
            const int bibase = islo ? 0 : 8;
            int bi = bibase;
#pragma unroll
            for (int i = 1; i < 8; ++i) {
                const bool gt = d[i] > best;
                best = gt ? d[i] : best;
                bi   = gt ? (bibase + i) : bi;
            }
            const float ob = hswap_f(best);
            const int   oi = hswap_i(bi);
            const float sL = islo ? best : ob;   // colors 0-7 candidate
            const int   iL = islo ? bi   : oi;
            const float sH = islo ? ob   : best; // colors 8-15 candidate
            const int   iH = islo ? oi   : bi;
            idxP[pt] = (sH > sL) ? iH : iL;
        }
        const int fi = islo ? idxP[0] : idxP[1];   // lane L -> pixel p
        const float o0 = SP[fi][0], o1 = SP[fi][1], o2 = SP[fi][2];
        __builtin_nontemporal_store(o0, out + base);
        __builtin_nontemporal_store(o1, out + base + HWC);
        __builtin_nontemporal_store(o2, out + base + 2*HWC);
    }
}

extern "C" void kernel_launch(void* const* d_in, const int* in_sizes, int n_in,
                              void* d_out, int out_size, void* d_ws, size_t ws_size,
                              hipStream_t stream) {
    const float* x  = (const float*)d_in[0];
    const float* W1 = (const float*)d_in[1];
    const float* b1 = (const float*)d_in[2];
    const float* W2 = (const float*)d_in[3];
    const float* b2 = (const float*)d_in[4];
    float* out = (float*)d_out;

    const int npix = in_sizes[0] / 3;   // 8,388,608 for 32x3x512x512
    const int tpb = 256;
    int blocks = 2048;
    // keep trip count uniform (EXEC must stay all-ones for WMMA)
    while (blocks > 1 && (npix % (blocks * tpb)) != 0) blocks >>= 1;

    cq_wmma_kernel<<<blocks, tpb, 0, stream>>>(x, W1, b1, W2, b2, out, npix);
}